// IOHMMClassification_40054865003141
// MI455X (gfx1250) — compile-verified
//
#include <hip/hip_runtime.h>

typedef float v2f __attribute__((ext_vector_type(2)));
typedef float v8f __attribute__((ext_vector_type(8)));

// -------- Kernel 1: row softmax of transition (64 rows) and output (16 rows) --------
__global__ __launch_bounds__(32) void iohmm_softmax_rows(
    const float* __restrict__ trans, const float* __restrict__ outp,
    float* __restrict__ Psm, float* __restrict__ Osm)
{
    const int row = blockIdx.x;
    const int lane = threadIdx.x;
    const float* src;
    float* dst;
    if (row < 64) { src = trans + row * 64; dst = Psm + row * 64; }
    else          { src = outp + (row - 64) * 64; dst = Osm + (row - 64) * 64; }

    float x0 = src[lane], x1 = src[lane + 32];
    float mx = fmaxf(x0, x1);
#pragma unroll
    for (int m = 16; m >= 1; m >>= 1) mx = fmaxf(mx, __shfl_xor(mx, m, 32));
    float s = __expf(x0 - mx) + __expf(x1 - mx);
#pragma unroll
    for (int m = 16; m >= 1; m >>= 1) s += __shfl_xor(s, m, 32);
    float inv = 1.0f / s;
    dst[lane]      = __expf(x0 - mx) * inv;
    dst[lane + 32] = __expf(x1 - mx) * inv;
}

// -------- Kernel 2: HMM forward recursion, one wave per 16-batch tile --------
__global__ __launch_bounds__(32) void iohmm_forward(
    const int* __restrict__ sent, const int* __restrict__ length,
    const float* __restrict__ emb, const float* __restrict__ Psm,
    const float* __restrict__ Osm, float* __restrict__ out, int maxlen)
{
    __shared__ __align__(16) float g[16][64];   // G[b][s'] staging (A operand source)
    __shared__ v2f bst[16][4][32];              // B operand: [kchunk][ntile][lane]

    const int lane  = threadIdx.x;
    const int n     = lane & 15;    // column within 16-wide tile (state / label)
    const int khalf = lane >> 4;    // which K-half this lane holds for A/B operands
    const int bt    = blockIdx.x * 16;

    // Stage B = P^T tiles into LDS in WMMA 4x16 layout.
    // B[k, nn] = Pt[4kc+k][16j+nn] = Psm[16j+nn][4kc+k]  -> contiguous float2 in a Psm row.
#pragma unroll
    for (int kc = 0; kc < 16; ++kc)
#pragma unroll
        for (int j = 0; j < 4; ++j)
            bst[kc][j][lane] = *(const v2f*)(Psm + (j * 16 + n) * 64 + kc * 4 + khalf * 2);

    // Lengths: lanes 0..15 own batch rows bt..bt+15.
    int lenv = (lane < 16) ? length[bt + lane] : 1;
    int tmax = lenv;
#pragma unroll
    for (int m = 16; m >= 1; m >>= 1) tmax = max(tmax, __shfl_xor(tmax, m, 32));
    int lm1[8];
#pragma unroll
    for (int r = 0; r < 8; ++r) lm1[r] = __shfl(lenv, r + 8 * khalf, 32) - 1;

    // Emission in C-matrix layout: ev[j][r] = log_softmax(emb[tok_b])[16j+n], b = r + 8*khalf.
    auto emis = [&](int t, float ev[4][8]) {
        int tok = (lane < 16) ? sent[(bt + lane) * maxlen + t] : 0;
        float raw[4][8];
#pragma unroll
        for (int r = 0; r < 8; ++r) {
            int tb = __shfl(tok, r + 8 * khalf, 32);
            const float* rowp = emb + (long)tb * 64;
#pragma unroll
            for (int j = 0; j < 4; ++j) raw[j][r] = rowp[j * 16 + n];
        }
#pragma unroll
        for (int r = 0; r < 8; ++r) {
            float mx = fmaxf(fmaxf(raw[0][r], raw[1][r]), fmaxf(raw[2][r], raw[3][r]));
#pragma unroll
            for (int m = 8; m >= 1; m >>= 1) mx = fmaxf(mx, __shfl_xor(mx, m, 32));
            float sum = 0.f;
#pragma unroll
            for (int j = 0; j < 4; ++j) sum += __expf(raw[j][r] - mx);
#pragma unroll
            for (int m = 8; m >= 1; m >>= 1) sum += __shfl_xor(sum, m, 32);
            float lse = __logf(sum) + mx;
#pragma unroll
            for (int j = 0; j < 4; ++j) ev[j][r] = raw[j][r] - lse;
        }
    };

    const v8f vzero = {0.f, 0.f, 0.f, 0.f, 0.f, 0.f, 0.f, 0.f};
    v8f c[4], h[4];

    // t = 0: f0 = emission[0]
    {
        float ev[4][8];
        emis(0, ev);
#pragma unroll
        for (int j = 0; j < 4; ++j)
#pragma unroll
            for (int r = 0; r < 8; ++r) { c[j][r] = ev[j][r]; h[j][r] = ev[j][r]; }
    }

    for (int t = 1; t < tmax; ++t) {
        // Row max M[b] over all 64 states (across 4 tiles + 16-lane half reduction).
        float M[8];
#pragma unroll
        for (int r = 0; r < 8; ++r) {
            float mx = fmaxf(fmaxf(c[0][r], c[1][r]), fmaxf(c[2][r], c[3][r]));
#pragma unroll
            for (int m = 8; m >= 1; m >>= 1) mx = fmaxf(mx, __shfl_xor(mx, m, 32));
            M[r] = mx;
        }
        // G = exp(f - M) -> LDS in [b][s'] order (A operand source).
#pragma unroll
        for (int r = 0; r < 8; ++r) {
            const int b = r + 8 * khalf;
#pragma unroll
            for (int j = 0; j < 4; ++j) g[b][j * 16 + n] = __expf(c[j][r] - M[r]);
        }
        // NF = G * P^T via f32 WMMA: 16 K-chunks of 4, 4 N-tiles of 16 states.
        v8f acc[4] = {vzero, vzero, vzero, vzero};
#pragma unroll
        for (int kc = 0; kc < 16; ++kc) {
            v2f a = *(const v2f*)&g[n][kc * 4 + 2 * khalf];
#pragma unroll
            for (int j = 0; j < 4; ++j)
                acc[j] = __builtin_amdgcn_wmma_f32_16x16x4_f32(
                    false, a, false, bst[kc][j][lane], (short)0, acc[j], false, false);
        }
        float ev[4][8];
        emis(t, ev);
#pragma unroll
        for (int j = 0; j < 4; ++j)
#pragma unroll
            for (int r = 0; r < 8; ++r) {
                float nf = __logf(acc[j][r]) + M[r] + ev[j][r];
                c[j][r] = nf;
                if (t == lm1[r]) h[j][r] = nf;   // capture hidden at length-1
            }
    }

    // Final: score[b,l] = log( sum_s softmax(output)[l,s] * exp(h[b,s]-Mh[b]) ) + Mh[b]
    float Mh[8];
#pragma unroll
    for (int r = 0; r < 8; ++r) {
        float mx = fmaxf(fmaxf(h[0][r], h[1][r]), fmaxf(h[2][r], h[3][r]));
#pragma unroll
        for (int m = 8; m >= 1; m >>= 1) mx = fmaxf(mx, __shfl_xor(mx, m, 32));
        Mh[r] = mx;
    }
#pragma unroll
    for (int r = 0; r < 8; ++r) {
        const int b = r + 8 * khalf;
#pragma unroll
        for (int j = 0; j < 4; ++j) g[b][j * 16 + n] = __expf(h[j][r] - Mh[r]);
    }
    v8f accs = vzero;
#pragma unroll
    for (int kc = 0; kc < 16; ++kc) {
        v2f a  = *(const v2f*)&g[n][kc * 4 + 2 * khalf];
        v2f bo = *(const v2f*)(Osm + n * 64 + kc * 4 + 2 * khalf);  // B[k,l]=Osm[l][k]
        accs = __builtin_amdgcn_wmma_f32_16x16x4_f32(
            false, a, false, bo, (short)0, accs, false, false);
    }
#pragma unroll
    for (int r = 0; r < 8; ++r)
        out[(bt + r + 8 * khalf) * 16 + n] = __logf(accs[r]) + Mh[r];
}

extern "C" void kernel_launch(void* const* d_in, const int* in_sizes, int n_in,
                              void* d_out, int out_size, void* d_ws, size_t ws_size,
                              hipStream_t stream) {
    const int*   sent   = (const int*)d_in[0];
    const int*   length = (const int*)d_in[1];
    const float* emb    = (const float*)d_in[2];
    const float* trans  = (const float*)d_in[3];
    const float* outp   = (const float*)d_in[4];
    float* out = (float*)d_out;

    float* Psm = (float*)d_ws;        // 64*64 floats
    float* Osm = Psm + 64 * 64;       // 16*64 floats  (20 KB total workspace)

    const int batch  = in_sizes[1];               // 512
    const int maxlen = in_sizes[0] / batch;       // 512

    iohmm_softmax_rows<<<80, 32, 0, stream>>>(trans, outp, Psm, Osm);
    iohmm_forward<<<batch / 16, 32, 0, stream>>>(sent, length, emb, Psm, Osm, out, maxlen);
}